// HeatmapAlignmentLoss_41008347742310
// MI455X (gfx1250) — compile-verified
//
#include <hip/hip_runtime.h>
#include <cmath>
#include <cstdint>

// ---------------------------------------------------------------------------
// HeatmapAlignmentLoss for MI455X (gfx1250).
//
// Roofline: 151 MB of input @ 23.3 TB/s => ~6.5 us floor; only ~2.8 GFLOP of
// math => memory-bound. So: no WMMA (banded 11-tap conv as matmul would only
// inflate FLOPs), and use the CDNA5 Tensor Data Mover (tensor_load_to_lds,
// TENSORcnt) to DMA each 96x96 patch plane into LDS without VGPR staging.
// SSIM window is separable: vertical 11-row BOX (all-ones) x horizontal
// 11-tap gaussian (sum 1)  -> streaming column sums + register-reused FIR.
// ---------------------------------------------------------------------------

#define NPATCH 2048
#define HH 96
#define WW 96
#define NPIX (HH * WW)
#define NTHR 96          // 3 waves (wave32)
#define NBAND 4
#define BANDROWS 24      // 96 / 4
#define COLS_PER_THR 4
#define THR_PER_BAND 24  // 96 cols / 4
#define CSROW 112        // 8 left guard + 96 data + 8 right guard (16B-aligned f4)
#define SSIM_C1 1.0e-4f
#define SSIM_C2 9.0e-4f

struct GaussW { float w[11]; };

typedef unsigned int v4u __attribute__((ext_vector_type(4)));
typedef int          v8i __attribute__((ext_vector_type(8)));
typedef int          v4i __attribute__((ext_vector_type(4)));

// Build a 1-D TDM descriptor (D#) per CDNA5 ISA ch.8 and issue
// tensor_load_to_lds: contiguous n f32 elements global -> LDS.
// This toolchain exposes the 6-arg builtin:
//   (uint32x4 g0, int32x8 g1, int32x4 g2, int32x4 g3, int32x8 extra, i32 cpol)
__device__ __forceinline__ void tdm_load_f32_1d(const void* gsrc,
                                                uint32_t lds_byte_off,
                                                uint32_t n) {
  uint64_t ga = (uint64_t)(uintptr_t)gsrc;
  // Group 0 (128b): count=1 | lds_addr | global_addr[56:0] | type=2
  v4u g0 = { 1u,
             lds_byte_off,
             (uint32_t)ga,
             (uint32_t)(ga >> 32) | 0x80000000u };
  // Group 1 (256b): data_size=2 (4B); tensor_dim0=n; tensor_dim1=1;
  // tile_dim0=n; tile_dim1=1; tensor_dim0_stride=n; tensor_dim1_stride=n.
  v8i g1 = { (int)(2u << 16),                    // workgroup_mask=0, data_size=4B
             (int)((n & 0xFFFFu) << 16),         // tensor_dim0 lo16  (bits 63:48)
             (int)((n >> 16) | (1u << 16)),      // tensor_dim0 hi16 | tensor_dim1=1
             (int)((n & 0xFFFFu) << 16),         // tile_dim0 (bits 127:112)
             1,                                  // tile_dim1=1, tile_dim2=0
             (int)n,                             // tensor_dim0_stride lo32
             (int)((n & 0xFFFFu) << 16),         // stride0 hi16(=0) | stride1 lo16
             (int)(n >> 16) };                   // stride1 hi
  v4i z4 = { 0, 0, 0, 0 };                       // groups 2/3 unused (<=2D)
  v8i z8 = { 0, 0, 0, 0, 0, 0, 0, 0 };
  __builtin_amdgcn_tensor_load_to_lds(g0, g1, z4, z4, z8, 0);
}

__device__ __forceinline__ void accum_cols(float col[5][COLS_PER_THR],
                                           const float* __restrict__ sp,
                                           const float* __restrict__ st,
                                           int r, int x0, float sgn) {
  float4 pv = *(const float4*)(sp + r * WW + x0);
  float4 tv = *(const float4*)(st + r * WW + x0);
  float ps[4] = { pv.x, pv.y, pv.z, pv.w };
  float ts[4] = { tv.x, tv.y, tv.z, tv.w };
#pragma unroll
  for (int j = 0; j < 4; ++j) {
    float p = ps[j], t = ts[j];
    col[0][j] = fmaf(sgn, p,     col[0][j]);
    col[1][j] = fmaf(sgn, t,     col[1][j]);
    col[2][j] = fmaf(sgn, p * p, col[2][j]);
    col[3][j] = fmaf(sgn, t * t, col[3][j]);
    col[4][j] = fmaf(sgn, p * t, col[4][j]);
  }
}

__global__ void __launch_bounds__(NTHR)
heatmap_patch_kernel(const float* __restrict__ pred,
                     const float* __restrict__ tgt,
                     const unsigned char* __restrict__ mask,
                     float* __restrict__ loss_out,
                     float* __restrict__ valid_out,
                     GaussW gw) {
  extern __shared__ float smem[];
  float* sp  = smem;                       // [NPIX]   pred plane
  float* st  = smem + NPIX;                // [NPIX]   target plane
  float* cs  = smem + 2 * NPIX;            // [NBAND][5][CSROW] column sums
  float* red = cs + NBAND * 5 * CSROW;     // [3 * NTHR] reduction scratch

  const int tid   = threadIdx.x;
  const int patch = blockIdx.x;

  // ---- TDM DMA: both 36KB planes global -> LDS (issued once by wave 0) ----
  if (tid == 0) {
    uint32_t lds0 = (uint32_t)(uintptr_t)(void*)smem;  // low 32 bits = LDS offset
    tdm_load_f32_1d(pred + (size_t)patch * NPIX, lds0,             NPIX);
    tdm_load_f32_1d(tgt  + (size_t)patch * NPIX, lds0 + NPIX * 4u, NPIX);
  }
  __builtin_amdgcn_s_wait_tensorcnt(0);

  // Zero the horizontal guard columns of cs while the DMA is in flight.
  for (int i = tid; i < NBAND * 5 * CSROW; i += NTHR) {
    int c = i % CSROW;
    if (c < 8 || c >= 104) cs[i] = 0.f;
  }
  __syncthreads();   // DMA complete (wave0 waited) + guards visible

  // ---- BCE + target sum (one strided pass, lane-consecutive = no conflicts)
  float bce = 0.f, tsum = 0.f;
  for (int i = tid; i < NPIX; i += NTHR) {
    float p = sp[i], t = st[i];
    float lp = fmaxf(logf(p), -100.f);
    float lq = fmaxf(log1pf(-p), -100.f);
    bce -= t * lp + (1.f - t) * lq;
    tsum += t;
  }

  // ---- SSIM: 4 row-bands x 24 threads x 4 columns/thread ----
  const int band = tid / THR_PER_BAND;        // 0..3 -> rows [24*band, 24*band+23]
  const int q    = tid % THR_PER_BAND;
  const int x0   = q * COLS_PER_THR;          // 16B aligned column base
  const int y0   = band * BANDROWS;
  float* csb = cs + band * 5 * CSROW;

  // Column box-sums over the 11-row window (zero-padded), kept in registers.
  float col[5][COLS_PER_THR];
#pragma unroll
  for (int c = 0; c < 5; ++c)
#pragma unroll
    for (int j = 0; j < COLS_PER_THR; ++j) col[c][j] = 0.f;
  {
    int rlo = (y0 - 5 < 0) ? 0 : y0 - 5;
    int rhi = (y0 + 5 > HH - 1) ? HH - 1 : y0 + 5;
    for (int r = rlo; r <= rhi; ++r) accum_cols(col, sp, st, r, x0, 1.f);
  }

  float ssim_acc = 0.f;
  for (int s = 0; s < BANDROWS; ++s) {
    int y = y0 + s;
    if (s > 0) {                           // slide window: +row y+5, -row y-6
      int rin = y + 5, rout = y - 6;
      if (rin < HH)  accum_cols(col, sp, st, rin,  x0,  1.f);
      if (rout >= 0) accum_cols(col, sp, st, rout, x0, -1.f);
    }
    // Publish column sums (float4 stores -> conflict-free).
#pragma unroll
    for (int c = 0; c < 5; ++c) {
      float4 v = make_float4(col[c][0], col[c][1], col[c][2], col[c][3]);
      *(float4*)(csb + c * CSROW + 8 + x0) = v;
    }
    __syncthreads();

    // Horizontal 11-tap gaussian: 5 aligned float4 loads cover cols
    // [x0-8 .. x0+11] of the guarded row; reuse across 4 outputs.
    float conv[5][COLS_PER_THR];
#pragma unroll
    for (int c = 0; c < 5; ++c) {
      const float* base = csb + c * CSROW + x0;  // index x0 == data col x0-8
      float l[20];
#pragma unroll
      for (int u = 0; u < 5; ++u) {
        float4 v = *(const float4*)(base + 4 * u);
        l[4 * u + 0] = v.x; l[4 * u + 1] = v.y;
        l[4 * u + 2] = v.z; l[4 * u + 3] = v.w;
      }
#pragma unroll
      for (int m = 0; m < COLS_PER_THR; ++m) {
        float a = 0.f;
#pragma unroll
        for (int k = 0; k < 11; ++k) a = fmaf(gw.w[k], l[3 + m + k], a);
        conv[c][m] = a;
      }
    }
#pragma unroll
    for (int m = 0; m < COLS_PER_THR; ++m) {
      float mu1 = conv[0][m], mu2 = conv[1][m];
      float e11 = conv[2][m], e22 = conv[3][m], e12 = conv[4][m];
      float mu1s = mu1 * mu1, mu2s = mu2 * mu2, mu12 = mu1 * mu2;
      float s1 = e11 - mu1s, s2 = e22 - mu2s, s12 = e12 - mu12;
      float num = (2.f * mu12 + SSIM_C1) * (2.f * s12 + SSIM_C2);
      float den = (mu1s + mu2s + SSIM_C1) * (s1 + s2 + SSIM_C2);
      ssim_acc += num / den;
    }
    __syncthreads();   // WAR: next iteration overwrites csb
  }

  // ---- Deterministic per-patch reduction (fixed order) ----
  red[tid] = bce;
  red[NTHR + tid] = tsum;
  red[2 * NTHR + tid] = ssim_acc;
  __syncthreads();
  if (tid == 0) {
    float bT = 0.f, tT = 0.f, sT = 0.f;
    for (int i = 0; i < NTHR; ++i) {
      bT += red[i]; tT += red[NTHR + i]; sT += red[2 * NTHR + i];
    }
    float bce_mean  = bT * (1.f / (float)NPIX);
    float ssim_mean = sT * (1.f / (float)NPIX);
    float loss = 0.5f * bce_mean + 0.5f * (1.f - ssim_mean);
    bool valid = (mask[patch] != 0) && (tT >= 1e-6f);
    loss_out[patch]  = valid ? loss : 0.f;
    valid_out[patch] = valid ? 1.f : 0.f;
  }
}

__global__ void __launch_bounds__(256)
final_reduce_kernel(const float* __restrict__ loss,
                    const float* __restrict__ valid,
                    float* __restrict__ out) {
  __shared__ float sl[256], sc[256];
  int tid = threadIdx.x;
  float a = 0.f, c = 0.f;
  for (int i = tid; i < NPATCH; i += 256) { a += loss[i]; c += valid[i]; }
  sl[tid] = a; sc[tid] = c;
  __syncthreads();
  for (int off = 128; off > 0; off >>= 1) {
    if (tid < off) { sl[tid] += sl[tid + off]; sc[tid] += sc[tid + off]; }
    __syncthreads();
  }
  if (tid == 0) {
    float cnt = sc[0];
    out[0] = (cnt > 0.f) ? (sl[0] / cnt) : 0.f;
  }
}

extern "C" void kernel_launch(void* const* d_in, const int* in_sizes, int n_in,
                              void* d_out, int out_size, void* d_ws, size_t ws_size,
                              hipStream_t stream) {
  const float* pred = (const float*)d_in[0];
  const float* tgt  = (const float*)d_in[1];
  const unsigned char* mask = (const unsigned char*)d_in[2];  // jnp bool = 1 byte

  float* loss_arr  = (float*)d_ws;           // [2048]
  float* valid_arr = loss_arr + NPATCH;      // [2048]

  GaussW gw;
  {
    double g[11], s = 0.0;
    for (int j = 0; j < 11; ++j) { double x = j - 5; g[j] = exp(-(x * x) / 4.5); s += g[j]; }
    for (int j = 0; j < 11; ++j) gw.w[j] = (float)(g[j] / s);
  }

  const size_t smem_bytes =
      (size_t)(2 * NPIX + NBAND * 5 * CSROW + 3 * NTHR) * sizeof(float); // ~84 KB
  (void)hipFuncSetAttribute((const void*)heatmap_patch_kernel,
                            hipFuncAttributeMaxDynamicSharedMemorySize,
                            (int)smem_bytes);

  heatmap_patch_kernel<<<NPATCH, NTHR, smem_bytes, stream>>>(
      pred, tgt, mask, loss_arr, valid_arr, gw);
  final_reduce_kernel<<<1, 256, 0, stream>>>(loss_arr, valid_arr, (float*)d_out);
}